// GraphLayer_17970143167130
// MI455X (gfx1250) — compile-verified
//
#include <hip/hip_runtime.h>
#include <hip/hip_bf16.h>

// Problem constants (from reference)
#define BATCH  4
#define NNODE  50000
#define NEDGE  800000
#define FEAT   128
#define MROWS  (BATCH * NNODE)          // 200000 rows total
#define NTILES (MROWS / 16)             // 12500 16-row WMMA tiles (exact)
#define BN_EPS 1e-5f
#define SLOPE  0.01f

typedef __attribute__((ext_vector_type(2))) float v2f;
typedef __attribute__((ext_vector_type(8))) float v8f;

#if __has_builtin(__builtin_amdgcn_global_load_async_to_lds_b128)
#define HAS_ASYNC_LDS 1
// Builtin signature (from hipcc diagnostic + LLVM intrinsic):
//   void(v4i addrspace(1)*, v4i addrspace(3)*, i32 imm offset, i32 imm cpol)
typedef int v4i __attribute__((vector_size(16)));
typedef __attribute__((address_space(1))) v4i as1_v4i;
typedef __attribute__((address_space(3))) v4i as3_v4i;
#else
#define HAS_ASYNC_LDS 0
#endif

// ---------------------------------------------------------------------------
// Workspace layout (floats):
//   agg   : [MROWS*FEAT]  aggregated (A_norm @ x), 102.4 MB
//   deg   : [MROWS]       degree -> becomes dis = rsqrt(deg) in place
//   stats : [512]         [0..127]=sum  [128..255]=sumsq  [256..383]=scale
//                         [384..511]=shift
// ---------------------------------------------------------------------------
#define WS_AGG_OFF   0
#define WS_DEG_OFF   ((size_t)MROWS * FEAT)
#define WS_STAT_OFF  (WS_DEG_OFF + MROWS)

// ---- 1) init: zero agg, deg=1 (self-loop weight), zero stats --------------
__global__ void gcn_init(float* __restrict__ agg, float* __restrict__ deg,
                         float* __restrict__ stats) {
    const long i = (long)blockIdx.x * blockDim.x + threadIdx.x;
    if (i < (long)MROWS * FEAT) agg[i] = 0.0f;
    if (i < MROWS)              deg[i] = 1.0f;   // self loop, weight 1
    if (i < 512)                stats[i] = 0.0f;
}

// ---- 2) degree accumulation: deg[b, col[e]] += w[b, e] --------------------
__global__ void gcn_degree(const int* __restrict__ edges,
                           const float* __restrict__ wts,
                           float* __restrict__ deg) {
    const long t = (long)blockIdx.x * blockDim.x + threadIdx.x;
    if (t >= (long)BATCH * NEDGE) return;
    const int b = (int)(t / NEDGE);
    const int e = (int)(t % NEDGE);
    const int col = edges[(size_t)b * 2 * NEDGE + NEDGE + e];
    unsafeAtomicAdd(&deg[b * NNODE + col], wts[t]);
}

// ---- 3) deg -> dis = rsqrt(deg) in place ----------------------------------
__global__ void gcn_rsqrt(float* __restrict__ deg) {
    const long i = (long)blockIdx.x * blockDim.x + threadIdx.x;
    if (i >= MROWS) return;
    const float d = deg[i];
    deg[i] = d > 0.0f ? rsqrtf(d) : 0.0f;
}

// ---- 4) edge scatter: agg[b,col] += x[b,row] * norm  (1 wave per edge) ----
__global__ __launch_bounds__(256) void gcn_scatter(
    const float* __restrict__ x, const int* __restrict__ edges,
    const float* __restrict__ wts, const float* __restrict__ dis,
    float* __restrict__ agg) {
    const long slot = (long)blockIdx.x * 8 + (threadIdx.x >> 5);
    if (slot >= (long)BATCH * NEDGE) return;
    const int lane = threadIdx.x & 31;
    const int b = (int)(slot / NEDGE);
    const int e = (int)(slot % NEDGE);
    const int* eb = edges + (size_t)b * 2 * NEDGE;
    const int row = eb[e];
    const int col = eb[NEDGE + e];
    const float norm = dis[b * NNODE + row] * wts[slot] * dis[b * NNODE + col];
    const float4 v = *(const float4*)(x + ((size_t)(b * NNODE + row)) * FEAT + lane * 4);
    float* dst = agg + ((size_t)(b * NNODE + col)) * FEAT + lane * 4;
    unsafeAtomicAdd(dst + 0, v.x * norm);
    unsafeAtomicAdd(dst + 1, v.y * norm);
    unsafeAtomicAdd(dst + 2, v.z * norm);
    unsafeAtomicAdd(dst + 3, v.w * norm);
}

// ---- 5) WMMA GEMM: out = (agg + x*dis^2) @ W + b, fused BN partial sums ---
// Block = 256 threads = 8 waves; wave w owns 16-row tile (blockIdx*8 + w).
// A fragment (16x4 f32): lane L: row = L&15, k-pair = 2*(L>>4).
// B fragment (4x16 f32) mirrored, sourced from LDS-resident W (64 KB).
// C/D (16x16 f32): VGPR i holds rows i (lanes 0-15) and i+8 (lanes 16-31).
// Two independent accumulator chains halve the WMMA RAW critical path.
__global__ __launch_bounds__(256) void gcn_gemm_wmma(
    const float* __restrict__ agg, const float* __restrict__ x,
    const float* __restrict__ dis, const float* __restrict__ Wm,
    const float* __restrict__ bias, float* __restrict__ out,
    float* __restrict__ gsum, float* __restrict__ gsq) {
    __shared__ float lW[FEAT * FEAT];           // exactly 64 KB
#if HAS_ASYNC_LDS
    {
        // CDNA5 async global->LDS: no VGPR staging, tracked by ASYNCcnt.
        for (int i = threadIdx.x * 4; i < FEAT * FEAT; i += 256 * 4) {
            as1_v4i* gsrc = (as1_v4i*)(uintptr_t)(Wm + i);
            as3_v4i* ldst = (as3_v4i*)(unsigned)(uintptr_t)(&lW[i]);
            __builtin_amdgcn_global_load_async_to_lds_b128(gsrc, ldst, 0, 0);
        }
#if __has_builtin(__builtin_amdgcn_s_wait_asynccnt)
        __builtin_amdgcn_s_wait_asynccnt(0);
#else
        asm volatile("s_wait_asynccnt 0" ::: "memory");
#endif
    }
#else
    for (int i = threadIdx.x; i < FEAT * FEAT; i += 256) lW[i] = Wm[i];
#endif
    __syncthreads();

    const int lane = threadIdx.x & 31;
    const int wave = threadIdx.x >> 5;
    const long tile = (long)blockIdx.x * 8 + wave;
    if (tile >= NTILES) return;                 // wave-uniform guard

    const int rowBase = (int)tile * 16;
    const int m  = lane & 15;
    const int kb = (lane >> 4) * 2;             // 0 or 2
    const int r  = rowBase + m;
    const float d  = dis[r];
    const float d2 = d * d;                     // self-loop norm dis^2
    const float* ar = agg + (size_t)r * FEAT;
    const float* xr = x   + (size_t)r * FEAT;

    // Preload the wave's full 16x128 A strip as WMMA fragments (64 VGPRs).
    v2f areg[32];
#pragma unroll
    for (int k = 0; k < 32; ++k) {
        const int c = 4 * k + kb;
        v2f a;
        a.x = ar[c]     + xr[c]     * d2;       // fold self-loop message
        a.y = ar[c + 1] + xr[c + 1] * d2;
        areg[k] = a;
    }

#pragma unroll 1
    for (int ct = 0; ct < 8; ++ct) {            // 8 column tiles of 16
        const int n = ct * 16 + m;              // this lane's output column
        const float bb = bias[n];
        v8f acc0 = {bb, bb, bb, bb, bb, bb, bb, bb};
        v8f acc1 = {0.f, 0.f, 0.f, 0.f, 0.f, 0.f, 0.f, 0.f};
#pragma unroll
        for (int k = 0; k < 32; k += 2) {       // K = 32 * 4 = 128, 2 chains
            v2f bf0, bf1;
            bf0.x = lW[(4 * k + kb)     * FEAT + n];
            bf0.y = lW[(4 * k + kb + 1) * FEAT + n];
            bf1.x = lW[(4 * k + 4 + kb)     * FEAT + n];
            bf1.y = lW[(4 * k + 4 + kb + 1) * FEAT + n];
            acc0 = __builtin_amdgcn_wmma_f32_16x16x4_f32(
                false, areg[k],     false, bf0, (short)0, acc0, false, false);
            acc1 = __builtin_amdgcn_wmma_f32_16x16x4_f32(
                false, areg[k + 1], false, bf1, (short)0, acc1, false, false);
        }
        // Store + BN partial stats. Lane's column is fixed; rows i / i+8.
        const int rb = rowBase + (lane >> 4) * 8;
        float s = 0.0f, q = 0.0f;
#pragma unroll
        for (int i = 0; i < 8; ++i) {
            const float v = acc0[i] + acc1[i];
            out[(size_t)(rb + i) * FEAT + n] = v;
            s += v;
            q += v * v;
        }
        // lanes L and L+16 hold the same column -> pairwise reduce first
        s += __shfl_xor(s, 16, 32);
        q += __shfl_xor(q, 16, 32);
        if (lane < 16) {
            unsafeAtomicAdd(&gsum[n], s);
            unsafeAtomicAdd(&gsq[n], q);
        }
    }
}

// ---- 6) BN finalize: scale/shift per channel ------------------------------
__global__ void gcn_bn_finalize(const float* __restrict__ gamma,
                                const float* __restrict__ beta,
                                float* __restrict__ stats) {
    const int j = threadIdx.x;
    if (j >= FEAT) return;
    const float cnt  = (float)MROWS;
    const float mean = stats[j] / cnt;
    const float var  = stats[FEAT + j] / cnt - mean * mean;  // biased var
    const float sc   = gamma[j] * rsqrtf(var + BN_EPS);
    stats[256 + j] = sc;
    stats[384 + j] = beta[j] - mean * sc;
}

// ---- 7) BN apply + LeakyReLU in place -------------------------------------
__global__ void gcn_bn_apply(float* __restrict__ out,
                             const float* __restrict__ stats) {
    const long i = (long)blockIdx.x * blockDim.x + threadIdx.x;
    if (i >= (long)MROWS * FEAT) return;
    const int j = (int)(i & (FEAT - 1));
    const float v = out[i] * stats[256 + j] + stats[384 + j];
    out[i] = v > 0.0f ? v : SLOPE * v;
}

// ---------------------------------------------------------------------------
extern "C" void kernel_launch(void* const* d_in, const int* in_sizes, int n_in,
                              void* d_out, int out_size, void* d_ws, size_t ws_size,
                              hipStream_t stream) {
    const float* x      = (const float*)d_in[0];  // [B,N,128]
    const int*   edges  = (const int*)d_in[1];    // [B,2,E] (JAX x64-off -> int32)
    const float* wts    = (const float*)d_in[2];  // [B,E]
    const float* Wm     = (const float*)d_in[3];  // [128,128]
    const float* bias   = (const float*)d_in[4];  // [128]
    const float* gamma  = (const float*)d_in[5];  // [128]
    const float* beta   = (const float*)d_in[6];  // [128]
    float* out = (float*)d_out;                   // [B,N,128]

    float* ws    = (float*)d_ws;
    float* agg   = ws + WS_AGG_OFF;
    float* deg   = ws + WS_DEG_OFF;               // becomes dis after rsqrt
    float* stats = ws + WS_STAT_OFF;

    const long elems = (long)MROWS * FEAT;        // 25,600,000
    const int  ewB   = (int)((elems + 255) / 256);          // 100000
    const int  degB  = (int)(((long)BATCH * NEDGE + 255) / 256);  // 12500
    const int  rsB   = (MROWS + 255) / 256;                 // 782
    const int  scB   = (int)(((long)BATCH * NEDGE * 32 + 255) / 256); // 400000
    const int  gmB   = (NTILES + 7) / 8;                    // 1563

    gcn_init<<<ewB, 256, 0, stream>>>(agg, deg, stats);
    gcn_degree<<<degB, 256, 0, stream>>>(edges, wts, deg);
    gcn_rsqrt<<<rsB, 256, 0, stream>>>(deg);
    gcn_scatter<<<scB, 256, 0, stream>>>(x, edges, wts, deg, agg);
    gcn_gemm_wmma<<<gmB, 256, 0, stream>>>(agg, x, deg, Wm, bias, out,
                                           stats, stats + FEAT);
    gcn_bn_finalize<<<1, 128, 0, stream>>>(gamma, beta, stats);
    gcn_bn_apply<<<ewB, 256, 0, stream>>>(out, stats);
}